// GaussianVoxelizer_54769422959039
// MI455X (gfx1250) — compile-verified
//
#include <hip/hip_runtime.h>
#include <hip/hip_bf16.h>

#define NDIMS   16
#define DIMX    200
#define DIMY    200
#define DIMZ    16
#define TOTAL   (DIMX * DIMY * DIMZ)
#define VS      0.4f

typedef float v2f __attribute__((ext_vector_type(2)));
typedef float v8f __attribute__((ext_vector_type(8)));

// ---------------------------------------------------------------- zero d_out
__global__ void gv_zero_kernel(float* __restrict__ p, int n) {
    int i = blockIdx.x * blockDim.x + threadIdx.x;
    int stride = gridDim.x * blockDim.x;
    for (; i < n; i += stride) p[i] = 0.0f;
}

// decompose linear slot t -> (kx,ky,kz) within an (nx,ny,nz) bbox
__device__ __forceinline__ void gv_decomp(int t, int nyz, int nz,
                                          int& kx, int& ky, int& kz) {
    kx = t / nyz;
    const int rem = t - kx * nyz;
    ky = rem / nz;
    kz = rem - ky * nz;
}

// ------------------------------------------------- splat: one wave per gaussian
__global__ __launch_bounds__(256) void gv_splat_kernel(
    const float* __restrict__ means, const float* __restrict__ covs,
    const float* __restrict__ opac,  const float* __restrict__ feats,
    float* __restrict__ gdens, float* __restrict__ gfeat, int nGauss) {

    const int lane = threadIdx.x & 31;
    // readfirstlane -> wave-uniform gaussian index -> scalar (s_load) parameter fetch
    const int g = blockIdx.x * 8 + __builtin_amdgcn_readfirstlane((int)(threadIdx.x >> 5));
    if (g >= nGauss) return;            // wave-uniform: EXEC stays all-ones below

    const float mx = means[g * 3 + 0], my = means[g * 3 + 1], mz = means[g * 3 + 2];
    const float a  = covs[g * 9 + 0], b  = covs[g * 9 + 1], c_ = covs[g * 9 + 2];
    const float d_ = covs[g * 9 + 3], e  = covs[g * 9 + 4], f  = covs[g * 9 + 5];
    const float h_ = covs[g * 9 + 6], h  = covs[g * 9 + 7], i_ = covs[g * 9 + 8];
    const float op = opac[g];

    // analytic symmetric 3x3 inverse (adjugate / det)
    const float det = a * (e * i_ - f * h) - b * (d_ * i_ - f * h_) + c_ * (d_ * h - e * h_);
    const float id  = 1.0f / det;
    const float m00 = (e * i_ - f * h) * id,  m01 = (c_ * h - b * i_) * id,  m02 = (b * f - c_ * e) * id;
    const float m10 = (f * h_ - d_ * i_) * id, m11 = (a * i_ - c_ * h_) * id, m12 = (c_ * d_ - a * f) * id;
    const float m20 = (d_ * h - e * h_) * id, m21 = (b * h_ - a * h) * id,  m22 = (a * e - b * d_) * id;

    const float rx = 3.0f * sqrtf(a), ry = 3.0f * sqrtf(e), rz = 3.0f * sqrtf(i_);

    // bbox in voxel indices (trunc-toward-zero like the reference int32 cast)
    const int sx = max(0, (int)((mx - rx + 40.0f) / VS));
    const int sy = max(0, (int)((my - ry + 40.0f) / VS));
    const int sz = max(0, (int)((mz - rz +  1.0f) / VS));
    const int ex = min(DIMX, (int)((mx + rx + 40.0f) / VS) + 1);
    const int ey = min(DIMY, (int)((my + ry + 40.0f) / VS) + 1);
    const int ez = min(DIMZ, (int)((mz + rz +  1.0f) / VS) + 1);
    const int nx = max(ex - sx, 0), ny = max(ey - sy, 0), nz = max(ez - sz, 0);
    const int nyz = ny * nz;
    const int n   = nx * nyz;           // live voxels (typically 27..64, <= 216)
    if (n == 0) return;                 // wave-uniform

    const int sbase = sx * (DIMY * DIMZ) + sy * DIMZ + sz;   // uniform flat base

    // B operand for rank-1 WMMA: features in the K=0 row, lanes 0..15 hold N=0..15
    const float featN = feats[g * NDIMS + (lane & 15)];
    v2f bmat; bmat.x = (lane < 16) ? featN : 0.0f; bmat.y = 0.0f;

    const int nd    = lane & 15;
    const int mbase = (lane < 16) ? 0 : 8;     // row offset this lane scatters

    const int nIter = (n + 31) >> 5;
    for (int it = 0; it < nIter; ++it) {
        // ----- this lane's own voxel (slot t) -----
        const int  t     = (it << 5) + lane;
        const bool valid = (t < n);
        int kx, ky, kz;
        gv_decomp(valid ? t : 0, nyz, nz, kx, ky, kz);
        const int ix = sx + kx, iy = sy + ky, iz = sz + kz;

        const float px = (float)ix * VS - 40.0f - mx;
        const float py = (float)iy * VS - 40.0f - my;
        const float pz = (float)iz * VS -  1.0f - mz;
        const float q  = px * (m00 * px + m01 * py + m02 * pz)
                       + py * (m10 * px + m11 * py + m12 * pz)
                       + pz * (m20 * px + m21 * py + m22 * pz);

        const float dens = valid ? op * __expf(-0.5f * q) : 0.0f;
        const int   flat = valid ? (ix * (DIMY * DIMZ) + iy * DIMZ + iz) : 0;

        atomicAdd(&gdens[flat], dens);          // global_atomic_add_f32 -> L2

        // ---- feature contribution via V_WMMA_F32_16X16X4_F32 (exact f32 rank-1) ----
        // group0 = voxels of lanes 0..15 (own density); group1 = lanes 16..31 (one bpermute)
        v2f a0; a0.x = (lane < 16) ? dens : 0.0f; a0.y = 0.0f;   // A: 16x4, K=0 column
        const float densHi = __shfl(dens, lane | 16, 32);
        v2f a1; a1.x = (lane < 16) ? densHi : 0.0f; a1.y = 0.0f;
        v8f c0 = {}; v8f c1 = {};
        c0 = __builtin_amdgcn_wmma_f32_16x16x4_f32(false, a0, false, bmat,
                                                   (short)0, c0, false, false);
        c1 = __builtin_amdgcn_wmma_f32_16x16x4_f32(false, a1, false, bmat,
                                                   (short)0, c1, false, false);

        // D layout: lane L, VGPR r holds (M = r + 8*(L>=16), N = L&15).
        // Row M's voxel is slot it*32 + M (grp0) / it*32 + 16 + M (grp1): recompute
        // its flat index arithmetically (carry-increment), no cross-lane traffic.
        const int tb0 = (it << 5) + mbase;
        int kx0, ky0, kz0, kx1, ky1, kz1;
        gv_decomp(tb0,      nyz, nz, kx0, ky0, kz0);
        gv_decomp(tb0 + 16, nyz, nz, kx1, ky1, kz1);

        #pragma unroll
        for (int r = 0; r < 8; ++r) {
            int flat0 = sbase + kx0 * (DIMY * DIMZ) + ky0 * DIMZ + kz0;
            flat0 = min(flat0, TOTAL - 1);           // OOB slots add exact 0.0
            atomicAdd(&gfeat[flat0 * NDIMS + nd], c0[r]);

            int flat1 = sbase + kx1 * (DIMY * DIMZ) + ky1 * DIMZ + kz1;
            flat1 = min(flat1, TOTAL - 1);
            atomicAdd(&gfeat[flat1 * NDIMS + nd], c1[r]);

            // carry-increment (kz -> ky -> kx)
            kz0 += 1; { const bool w = (kz0 == nz); kz0 = w ? 0 : kz0; ky0 += w ? 1 : 0; }
            { const bool w = (ky0 == ny); ky0 = w ? 0 : ky0; kx0 += w ? 1 : 0; }
            kz1 += 1; { const bool w = (kz1 == nz); kz1 = w ? 0 : kz1; ky1 += w ? 1 : 0; }
            { const bool w = (ky1 == ny); ky1 = w ? 0 : ky1; kx1 += w ? 1 : 0; }
        }
    }
}

// ------------------------------------------------------------- normalize feats
__global__ void gv_norm_kernel(const float* __restrict__ gdens,
                               float* __restrict__ gfeat) {
    const int v = blockIdx.x * blockDim.x + threadIdx.x;
    if (v >= TOTAL) return;
    const float inv = 1.0f / fmaxf(gdens[v], 1e-6f);
    float4* row = (float4*)(gfeat + (size_t)v * NDIMS);
    #pragma unroll
    for (int q4 = 0; q4 < 4; ++q4) {
        float4 x = row[q4];
        if (q4 == 3) x.w += 1e-5f;      // reference seeds grid_feats[:, -1] = 1e-5
        x.x *= inv; x.y *= inv; x.z *= inv; x.w *= inv;
        row[q4] = x;
    }
}

extern "C" void kernel_launch(void* const* d_in, const int* in_sizes, int n_in,
                              void* d_out, int out_size, void* d_ws, size_t ws_size,
                              hipStream_t stream) {
    const float* means = (const float*)d_in[0];   // [N,3]
    const float* covs  = (const float*)d_in[1];   // [N,3,3]
    const float* opac  = (const float*)d_in[2];   // [N]
    const float* feats = (const float*)d_in[3];   // [N,16]
    const int nGauss = in_sizes[2];

    float* out   = (float*)d_out;
    float* gdens = out;                 // [TOTAL]
    float* gfeat = out + TOTAL;         // [TOTAL, 16]

    gv_zero_kernel<<<2048, 256, 0, stream>>>(out, out_size);

    const int nWaveBlocks = (nGauss + 7) / 8;     // 8 waves (gaussians) per block
    gv_splat_kernel<<<nWaveBlocks, 256, 0, stream>>>(means, covs, opac, feats,
                                                     gdens, gfeat, nGauss);

    gv_norm_kernel<<<(TOTAL + 255) / 256, 256, 0, stream>>>(gdens, gfeat);
}